// WeightedSamplesLoss_18451179503709
// MI455X (gfx1250) — compile-verified
//
#include <hip/hip_runtime.h>

typedef __attribute__((ext_vector_type(2))) float v2f;
typedef __attribute__((ext_vector_type(8))) float v8f;

#define NPTS 512
#define NEPS 59
#define BLOCK 256

// eps schedule: [16, exp(2ln4), exp(2ln4 + 2ln0.9), ..., (57 mid vals), 1e-4]
__device__ __forceinline__ float eps_at(int k) {
    if (k == 0)  return 16.0f;              // DIAMETER^P
    if (k == 58) return 1.0e-4f;            // BLUR^P
    const float start = 2.7725887222397811f;   // 2*ln(4)
    const float step  = -0.21072103131565256f; // 2*ln(0.9)
    return __expf(start + (float)(k - 1) * step);
}

// softmin over 512 columns for 512 rows, rows striped across 8 waves.
// s_ij = h'[j] + (row_i . col_j)/eps ;  out_i = sqrow[i] - eps*(max_j + log sum_j exp)
// h'[j] is injected through the WMMA C accumulator (broadcast per column).
__device__ __forceinline__ void softmin_pass(
    const float* __restrict__ r0, const float* __restrict__ r1,   // row coords (LDS)
    const float* __restrict__ c0, const float* __restrict__ c1,   // col coords (LDS)
    const float* __restrict__ hprime,                             // [512] LDS
    const float* __restrict__ sqrow,                              // 0.5*|row|^2 (LDS)
    float* __restrict__ outp,                                     // [512] LDS
    float eps, float inv_eps, int wave, int lane)
{
    const int l15 = lane & 15;
    for (int t = wave * 4; t < wave * 4 + 4; ++t) {
        // A fragment: 16x4 f32, K=0,1 in lanes 0-15 (vgpr0,1), K=2,3 pad = 0
        v2f afrag = {0.0f, 0.0f};
        if (lane < 16) {
            int row = t * 16 + l15;
            afrag.x = r0[row] * inv_eps;
            afrag.y = r1[row] * inv_eps;
        }
        float m[8], ssum[8];
#pragma unroll
        for (int r = 0; r < 8; ++r) { m[r] = -3.0e38f; ssum[r] = 0.0f; }

        // process 4 column tiles per step: 4 WMMAs, then 5 exps per 4 scores
        for (int ct = 0; ct < 32; ct += 4) {
            v8f d[4];
#pragma unroll
            for (int u = 0; u < 4; ++u) {
                int col = (ct + u) * 16 + l15;
                v2f bfrag = {0.0f, 0.0f};
                if (lane < 16) { bfrag.x = c0[col]; bfrag.y = c1[col]; }
                float hj = hprime[col];
                v8f cacc = {hj, hj, hj, hj, hj, hj, hj, hj};
                // D = A(16x4) x B(4x16) + C : s = dot/eps + h'  per 16x16 tile
                d[u] = __builtin_amdgcn_wmma_f32_16x16x4_f32(
                    false, afrag, false, bfrag, (short)0, cacc, false, false);
            }
#pragma unroll
            for (int r = 0; r < 8; ++r) {
                float s0 = d[0][r], s1 = d[1][r], s2 = d[2][r], s3 = d[3][r];
                float mn = fmaxf(fmaxf(m[r], s0),
                                 fmaxf(fmaxf(s1, s2), s3));
                ssum[r] = ssum[r] * __expf(m[r] - mn)
                        + __expf(s0 - mn) + __expf(s1 - mn)
                        + __expf(s2 - mn) + __expf(s3 - mn);
                m[r] = mn;
            }
        }
        // reduce across the 16 lanes of each half (xor 8,4,2,1 stays in-half)
#pragma unroll
        for (int off = 8; off >= 1; off >>= 1) {
#pragma unroll
            for (int r = 0; r < 8; ++r) {
                float mo = __shfl_xor(m[r], off, 32);
                float so = __shfl_xor(ssum[r], off, 32);
                float mn = fmaxf(m[r], mo);
                ssum[r]  = ssum[r] * __expf(m[r] - mn) + so * __expf(mo - mn);
                m[r]     = mn;
            }
        }
        // lanes 0-15 hold rows t*16+r, lanes 16-31 hold rows t*16+8+r
        if (l15 == 0) {
            int base = t * 16 + (lane >> 4) * 8;
#pragma unroll
            for (int r = 0; r < 8; ++r) {
                int i = base + r;
                outp[i] = sqrow[i] - eps * (m[r] + __logf(ssum[r]));
            }
        }
    }
}

__global__ __launch_bounds__(BLOCK)
void sinkhorn_batch_kernel(const float* __restrict__ a_g,
                           const float* __restrict__ x_g,
                           const float* __restrict__ b_g,
                           const float* __restrict__ y_g,
                           const float* __restrict__ w_g,
                           float* __restrict__ loss_ws)
{
    __shared__ float x0[NPTS], x1[NPTS], y0[NPTS], y1[NPTS];
    __shared__ float alog[NPTS], blog[NPTS], sqx[NPTS], sqy[NPTS];
    __shared__ float f[NPTS], g[NPTS], ft[NPTS], gt[NPTS];
    __shared__ float hx[NPTS], hy[NPTS];
    __shared__ float red[BLOCK];

    const int bidx = blockIdx.x;
    const int tid  = threadIdx.x;
    const int lane = tid & 31;
    const int wave = tid >> 5;

    for (int i = tid; i < NPTS; i += BLOCK) {
        float av = a_g[bidx * NPTS + i];
        float bv = b_g[bidx * NPTS + i];
        float xa = x_g[(bidx * NPTS + i) * 2 + 0];
        float xb = x_g[(bidx * NPTS + i) * 2 + 1];
        float ya = y_g[(bidx * NPTS + i) * 2 + 0];
        float yb = y_g[(bidx * NPTS + i) * 2 + 1];
        x0[i] = xa; x1[i] = xb; y0[i] = ya; y1[i] = yb;
        alog[i] = __logf(av); blog[i] = __logf(bv);
        sqx[i] = 0.5f * (xa * xa + xb * xb);
        sqy[i] = 0.5f * (ya * ya + yb * yb);
    }
    __syncthreads();

    // ---- init at eps0 = 16: f = softmin(C_xy, blog), g = softmin(C_yx, alog)
    {
        const float eps = 16.0f, inv = 1.0f / 16.0f;
        for (int i = tid; i < NPTS; i += BLOCK) {
            hx[i] = blog[i] - sqy[i] * inv;
            hy[i] = alog[i] - sqx[i] * inv;
        }
        __syncthreads();
        softmin_pass(x0, x1, y0, y1, hx, sqx, ft, eps, inv, wave, lane);
        softmin_pass(y0, y1, x0, x1, hy, sqy, gt, eps, inv, wave, lane);
        __syncthreads();
        for (int i = tid; i < NPTS; i += BLOCK) { f[i] = ft[i]; g[i] = gt[i]; }
        __syncthreads();
    }

    // ---- eps schedule scan (59 steps), averaged updates
    for (int k = 0; k < NEPS; ++k) {
        const float eps = eps_at(k);
        const float inv = 1.0f / eps;
        for (int i = tid; i < NPTS; i += BLOCK) {
            hx[i] = blog[i] + (g[i] - sqy[i]) * inv;
            hy[i] = alog[i] + (f[i] - sqx[i]) * inv;
        }
        __syncthreads();
        softmin_pass(x0, x1, y0, y1, hx, sqx, ft, eps, inv, wave, lane);
        softmin_pass(y0, y1, x0, x1, hy, sqy, gt, eps, inv, wave, lane);
        __syncthreads();
        for (int i = tid; i < NPTS; i += BLOCK) {
            f[i] = 0.5f * (f[i] + ft[i]);
            g[i] = 0.5f * (g[i] + gt[i]);
        }
        __syncthreads();
    }

    // ---- final extrapolation at eps_last = 1e-4 (results in ft/gt)
    {
        const float eps = 1.0e-4f, inv = 1.0e4f;
        for (int i = tid; i < NPTS; i += BLOCK) {
            hx[i] = blog[i] + (g[i] - sqy[i]) * inv;
            hy[i] = alog[i] + (f[i] - sqx[i]) * inv;
        }
        __syncthreads();
        softmin_pass(x0, x1, y0, y1, hx, sqx, ft, eps, inv, wave, lane);
        softmin_pass(y0, y1, x0, x1, hy, sqy, gt, eps, inv, wave, lane);
        __syncthreads();
    }

    // ---- loss_b = <a, f_fin> + <b, g_fin>, scaled by w[b]
    float partial = 0.0f;
    for (int i = tid; i < NPTS; i += BLOCK)
        partial += a_g[bidx * NPTS + i] * ft[i] + b_g[bidx * NPTS + i] * gt[i];
    red[tid] = partial;
    __syncthreads();
    for (int s = BLOCK / 2; s > 0; s >>= 1) {
        if (tid < s) red[tid] += red[tid + s];
        __syncthreads();
    }
    if (tid == 0) loss_ws[bidx] = red[0] * w_g[bidx];
}

__global__ void reduce64_kernel(const float* __restrict__ ws, float* __restrict__ out)
{
    __shared__ float r[64];
    int t = threadIdx.x;
    r[t] = ws[t];
    __syncthreads();
    for (int s = 32; s > 0; s >>= 1) {
        if (t < s) r[t] += r[t + s];
        __syncthreads();
    }
    if (t == 0) out[0] = r[0];
}

extern "C" void kernel_launch(void* const* d_in, const int* in_sizes, int n_in,
                              void* d_out, int out_size, void* d_ws, size_t ws_size,
                              hipStream_t stream)
{
    const float* a = (const float*)d_in[0]; // prob1  [64,512]
    const float* x = (const float*)d_in[1]; // coord1 [64,512,2]
    const float* b = (const float*)d_in[2]; // prob2  [64,512]
    const float* y = (const float*)d_in[3]; // coord2 [64,512,2]
    const float* w = (const float*)d_in[4]; // weights [64]
    float* out = (float*)d_out;
    float* ws  = (float*)d_ws;

    sinkhorn_batch_kernel<<<dim3(64), dim3(BLOCK), 0, stream>>>(a, x, b, y, w, ws);
    reduce64_kernel<<<dim3(1), dim3(64), 0, stream>>>(ws, out);
}